// IlluminationModel_429496729904
// MI455X (gfx1250) — compile-verified
//
#include <hip/hip_runtime.h>
#include <math.h>

#define N_TBINS 1024
#define KDIM    64
#define BETA    100.0f
#define SIGMA   10.0f
#define EPSN    1e-8f

typedef float v2f __attribute__((ext_vector_type(2)));
typedef float v8f __attribute__((ext_vector_type(8)));

__device__ __forceinline__ v8f wmma_f32x4(v2f a, v2f b, v8f c) {
  // D = A(16x4,f32) * B(4x16,f32) + C(16x16,f32)
  return __builtin_amdgcn_wmma_f32_16x16x4_f32(false, a, false, b, (short)0, c, false, false);
}

// ---------------------------------------------------------------------------
// prep1: x = relu(learnable); irf = normalized rolled gaussian; circular conv.
// irf_input[t] = (1/gsum) * sum_s x[s] * g[(t - s + mu) mod n]
// ---------------------------------------------------------------------------
__global__ __launch_bounds__(1024)
void prep_irf_kernel(const float* __restrict__ learnable,
                     float* __restrict__ irf_out,
                     float* __restrict__ irf_sum_out) {
  __shared__ float xs[N_TBINS];
  __shared__ float gs[N_TBINS];
  __shared__ float red[1024];
  const int t = threadIdx.x;
  const int mu = (N_TBINS - 1) / 2;

  float li = learnable[t];
  xs[t] = li > 0.0f ? li : 0.0f;
  float d = ((float)t - (float)mu) / SIGMA;
  float g = expf(-0.5f * d * d);
  gs[t] = g;
  red[t] = g;
  __syncthreads();
  for (int s = 512; s > 0; s >>= 1) {
    if (t < s) red[t] += red[t + s];
    __syncthreads();
  }
  const float gsum = red[0];
  __syncthreads();

  float acc = 0.0f;
  for (int s = 0; s < N_TBINS; ++s)
    acc = fmaf(xs[s], gs[(t - s + mu) & (N_TBINS - 1)], acc);
  const float irf_t = acc / gsum;
  irf_out[t] = irf_t;

  red[t] = irf_t;
  __syncthreads();
  for (int s = 512; s > 0; s >>= 1) {
    if (t < s) red[t] += red[t + s];
    __syncthreads();
  }
  if (t == 0) irf_sum_out[0] = red[0];
}

// ---------------------------------------------------------------------------
// prep2: corr_norm = zero-norm of cmat over K (per column n), stored
// pair-interleaved over k:  corrP[(k/2)*2048 + n*2 + (k&1)]
// cmat transposed & pair-interleaved over c: cmatP[(c/2)*128 + k*2 + (c&1)]
// so each WMMA B operand is one contiguous, coalesced 8-byte load.
// ---------------------------------------------------------------------------
__global__ __launch_bounds__(128)
void prep_corr_kernel(const float* __restrict__ cmat,
                      float* __restrict__ corrP,
                      float* __restrict__ cmatP) {
  const int n = blockIdx.x * blockDim.x + threadIdx.x;   // time bin 0..1023
  float s = 0.0f;
  for (int k = 0; k < KDIM; ++k) s += cmat[k * N_TBINS + n];
  const float mean = s * (1.0f / (float)KDIM);
  float ss = 0.0f;
  for (int k = 0; k < KDIM; ++k) {
    float dv = cmat[k * N_TBINS + n] - mean;
    ss = fmaf(dv, dv, ss);
  }
  const float inv = 1.0f / (sqrtf(ss) + EPSN);
  for (int k = 0; k < KDIM; ++k) {
    float v = cmat[k * N_TBINS + n];
    corrP[(k >> 1) * (N_TBINS * 2) + n * 2 + (k & 1)] = (v - mean) * inv;
    cmatP[(n >> 1) * (KDIM * 2) + k * 2 + (n & 1)]    = v;
  }
}

// ---------------------------------------------------------------------------
// main: per block = 16 batch rows, 4 waves (128 threads).
// noisy synth (pipelined eps) -> GEMM1 (WMMA f32 16x16x4) -> zero-norm
//   -> GEMM2 -> streaming softmax -> pred depth.
// ---------------------------------------------------------------------------
__global__ __launch_bounds__(128)
void illum_main_kernel(const int*   __restrict__ bins,
                       const float* __restrict__ pc,
                       const float* __restrict__ sbr,
                       const float* __restrict__ eps,
                       const float* __restrict__ irf,
                       const float* __restrict__ irf_sum,
                       const float* __restrict__ cmatP,
                       const float* __restrict__ corrP,
                       float* __restrict__ zncc_out,
                       float* __restrict__ pred_out) {
  __shared__ __align__(16) float irf_s[N_TBINS];      // 4 KB
  __shared__ __align__(16) float chunk[16 * 132];     // noisy tile chunk
  __shared__ __align__(16) float outn[16 * 68];       // GEMM1 out / input_norm
  __shared__ __align__(16) float stage[4 * 16 * 17];  // per-wave zncc staging
  __shared__ float s_scal[16], s_amb[16];
  __shared__ int   s_shift[16];
  __shared__ float wm[8 * 16], we[8 * 16], wt[8 * 16];

  const int tid  = threadIdx.x;
  const int w    = tid >> 5;       // wave id 0..3
  const int lane = tid & 31;
  const int l16  = lane & 15;
  const int hi   = lane >> 4;      // 0: K 0..1 half, 1: K 2..3 half
  const int koff = hi * 2;
  const int b0   = blockIdx.x * 16;

  for (int i = tid; i < N_TBINS; i += 128) irf_s[i] = irf[i];
  if (tid < 16) {
    const int r = tid;
    const float p  = pc[b0 + r];
    const float sb = sbr[b0 + r];
    s_shift[r] = bins[b0 + r] & (N_TBINS - 1);
    s_amb[r]   = p / (sb * (float)N_TBINS);
    s_scal[r]  = p / irf_sum[0];
  }
  __syncthreads();

  // ---- GEMM1: out[16 x 64] = noisy[16 x 1024] * cmat^T, chunked over c ----
  // Software-pipelined eps: chunk ch+1's eps loads fly during chunk ch's WMMAs.
  float ereg[16];
  #pragma unroll
  for (int j = 0; j < 16; ++j)
    ereg[j] = eps[(size_t)(b0 + j) * N_TBINS + tid];

  v8f acc = {};
  for (int ch = 0; ch < 8; ++ch) {
    const int cbase = ch * 128;
    #pragma unroll
    for (int j = 0; j < 16; ++j) {
      const int c = cbase + tid;
      const float shifted = irf_s[(c - s_shift[j]) & (N_TBINS - 1)];
      const float v = fmaf(shifted, s_scal[j], s_amb[j]);
      chunk[j * 132 + tid] = fmaf(sqrtf(v), ereg[j], v);
    }
    __syncthreads();

    if (ch < 7) {
      #pragma unroll
      for (int j = 0; j < 16; ++j) {
        ereg[j] = eps[(size_t)(b0 + j) * N_TBINS + cbase + 128 + tid];
        if (ch < 6)
          __builtin_prefetch(&eps[(size_t)(b0 + j) * N_TBINS + cbase + 256 + tid], 0, 0);
      }
    }

    const int ncol = w * 16 + l16;       // output k-column for this wave
    #pragma unroll 8
    for (int cc = 0; cc < 128; cc += 4) {
      v2f a = *(const v2f*)&chunk[l16 * 132 + cc + koff];   // A[m=l16][k=cc+koff..+1]
      const int kk = cbase + cc + koff;                     // even
      v2f b = *(const v2f*)&cmatP[(size_t)(kk >> 1) * (KDIM * 2) + ncol * 2];
      acc = wmma_f32x4(a, b, acc);
    }
    __syncthreads();
  }

  // ---- store 16x16 tile, zero-norm rows over all 64 k ----
  {
    const int col = w * 16 + l16;
    #pragma unroll
    for (int i = 0; i < 8; ++i)
      outn[(i + hi * 8) * 68 + col] = acc[i];
  }
  __syncthreads();
  if (tid < 16) {
    float s = 0.0f;
    for (int k = 0; k < KDIM; ++k) s += outn[tid * 68 + k];
    const float mean = s * (1.0f / (float)KDIM);
    float ss = 0.0f;
    for (int k = 0; k < KDIM; ++k) {
      float dv = outn[tid * 68 + k] - mean;
      ss = fmaf(dv, dv, ss);
    }
    const float inv = 1.0f / (sqrtf(ss) + EPSN);
    for (int k = 0; k < KDIM; ++k)
      outn[tid * 68 + k] = (outn[tid * 68 + k] - mean) * inv;
  }
  __syncthreads();

  // ---- GEMM2 + streaming softmax: zncc[16 x 1024], each wave owns 256 bins.
  // All 32 lanes keep stats: lane (l16,hi) scans 8 columns of row l16.
  float m = -INFINITY, sE = 0.0f, sET = 0.0f;
  for (int t = 0; t < 16; ++t) {
    const int nb = w * 256 + t * 16;
    v8f c2 = {};
    #pragma unroll
    for (int k0 = 0; k0 < KDIM; k0 += 4) {
      v2f a = *(const v2f*)&outn[l16 * 68 + k0 + koff];
      const int kk = k0 + koff;                             // even
      v2f b = *(const v2f*)&corrP[(size_t)(kk >> 1) * (N_TBINS * 2) + (nb + l16) * 2];
      c2 = wmma_f32x4(a, b, c2);
    }
    #pragma unroll
    for (int i = 0; i < 8; ++i) {
      const int row = i + hi * 8;
      const float z = c2[i];
      zncc_out[(size_t)(b0 + row) * N_TBINS + nb + l16] = z;
      stage[(w * 16 + row) * 17 + l16] = z;
    }
    __syncthreads();
    #pragma unroll
    for (int c = 0; c < 8; ++c) {
      const int col = hi * 8 + c;
      const float z = stage[(w * 16 + l16) * 17 + col];
      if (z > m) {
        const float f = expf(BETA * (m - z));
        sE *= f; sET *= f; m = z;
      }
      const float e = expf(BETA * (z - m));
      sE += e;
      sET = fmaf(e, (float)(nb + col), sET);
    }
    __syncthreads();
  }

  {
    const int g = w * 2 + hi;            // 8 stat groups per row
    wm[g * 16 + l16] = m;
    we[g * 16 + l16] = sE;
    wt[g * 16 + l16] = sET;
  }
  __syncthreads();
  if (tid < 16) {
    float M = wm[tid];
    for (int gg = 1; gg < 8; ++gg) M = fmaxf(M, wm[gg * 16 + tid]);
    float S = 0.0f, T = 0.0f;
    for (int gg = 0; gg < 8; ++gg) {
      const float f = expf(BETA * (wm[gg * 16 + tid] - M));
      S = fmaf(we[gg * 16 + tid], f, S);
      T = fmaf(wt[gg * 16 + tid], f, T);
    }
    pred_out[b0 + tid] = T / S;
  }
}

// ---------------------------------------------------------------------------
extern "C" void kernel_launch(void* const* d_in, const int* in_sizes, int n_in,
                              void* d_out, int out_size, void* d_ws, size_t ws_size,
                              hipStream_t stream) {
  const int*   bins  = (const int*)d_in[0];
  const float* pcnt  = (const float*)d_in[1];
  const float* sbrs  = (const float*)d_in[2];
  const float* learn = (const float*)d_in[3];
  const float* cmat  = (const float*)d_in[4];
  const float* eps   = (const float*)d_in[5];
  const int B = in_sizes[0];

  float* ws      = (float*)d_ws;
  float* irf     = ws;                    // 1024 f32
  float* irf_sum = ws + 1024;             // 1 f32
  float* cmatP   = ws + 1040;             // 1024*64 f32, pair-interleaved over c
  float* corrP   = cmatP + N_TBINS * KDIM;// 64*1024 f32, pair-interleaved over k

  float* zncc = (float*)d_out;                   // (B,1024,1) flat
  float* pred = zncc + (size_t)B * N_TBINS;      // (B,)

  prep_irf_kernel<<<1, 1024, 0, stream>>>(learn, irf, irf_sum);
  prep_corr_kernel<<<N_TBINS / 128, 128, 0, stream>>>(cmat, corrP, cmatP);
  illum_main_kernel<<<B / 16, 128, 0, stream>>>(bins, pcnt, sbrs, eps, irf,
                                                irf_sum, cmatP, corrP, zncc, pred);
}